// BEVFeatureExtractor_43508018709082
// MI455X (gfx1250) — compile-verified
//
#include <hip/hip_runtime.h>

// ---------------------------------------------------------------------------
// BEV feature extractor (CenterPoint-style bilinear gather), gfx1250.
//
// out[b,n,c] = wa*F[b,c,y0,x0] + wb*F[b,c,y1,x0] + wc*F[b,c,y0,x1] + wd*F[b,c,y1,x1]
//
// Cast as D = A(16x4) x B(4x16) per V_WMMA_F32_16X16X4_F32:
//   A rows all = (wa,wb,wc,wd)   (K = the 4 bilinear taps)
//   B[k][j]    = tap k of channel j
// One wave handles one point; 2 WMMAs per iteration cover 32 channels.
// ---------------------------------------------------------------------------

typedef __attribute__((ext_vector_type(2))) float v2f;
typedef __attribute__((ext_vector_type(8))) float v8f;

#define HAS_WMMA_F32_4 __has_builtin(__builtin_amdgcn_wmma_f32_16x16x4_f32)

namespace {
constexpr int   kH = 188, kW = 188, kC = 256, kN = 500, kB = 4;
constexpr float kPCX = -75.2f, kPCY = -75.2f, kVox = 0.1f, kStride = 8.0f;
}

__device__ __forceinline__ int iclamp(int v, int lo, int hi) {
  return v < lo ? lo : (v > hi ? hi : v);
}

__global__ __launch_bounds__(256) void bev_bilinear_wmma(
    const float* __restrict__ feats,    // (B, C, H, W)
    const float* __restrict__ centers,  // (B, N, 3)
    float* __restrict__ out)            // (B, N, C)
{
  const int lane = threadIdx.x & 31;
  const int wave = threadIdx.x >> 5;
  const int p    = blockIdx.x * 8 + wave;   // flat point id, 0..B*N-1 (exact)
  const int b    = p / kN;

  // --- per-point coords (uniform across the wave) ---
  const float cx = centers[(size_t)p * 3 + 0];
  const float cy = centers[(size_t)p * 3 + 1];
  // Match reference exactly: two successive divisions.
  float x = (cx - kPCX) / kVox; x = x / kStride;
  float y = (cy - kPCY) / kVox; y = y / kStride;

  const int fx = (int)floorf(x);
  const int fy = (int)floorf(y);
  const int x0 = iclamp(fx,     0, kW - 1);
  const int x1 = iclamp(fx + 1, 0, kW - 1);
  const int y0 = iclamp(fy,     0, kH - 1);
  const int y1 = iclamp(fy + 1, 0, kH - 1);

  // Weights computed from CLAMPED indices (matches reference).
  const float x0f = (float)x0, x1f = (float)x1;
  const float y0f = (float)y0, y1f = (float)y1;
  const float wa = (x1f - x) * (y1f - y);   // tap (y0,x0)
  const float wb = (x1f - x) * (y - y0f);   // tap (y1,x0)
  const float wc = (x - x0f) * (y1f - y);   // tap (y0,x1)
  const float wd = (x - x0f) * (y - y0f);   // tap (y1,x1)

  // --- WMMA operand layouts (wave32) ---
  // A 16x4 f32: lanes 0-15 -> K=0 (v0), K=1 (v1); lanes 16-31 -> K=2,3.
  const int half = lane >> 4;   // 0: x0 taps (K0/K1), 1: x1 taps (K2/K3)
  const int cl   = lane & 15;   // channel column within the 16-wide tile
  v2f A;
  A.x = half ? wc : wa;
  A.y = half ? wd : wb;

  // B 4x16 f32: lane%16 = column (channel); lanes 0-15 hold K=0 (v0), K=1 (v1)
  // = taps at x0 rows y0/y1; lanes 16-31 hold K=2,3 = taps at x1 rows y0/y1.
  const int    xt    = half ? x1 : x0;
  const size_t plane = (size_t)kH * kW;                 // channel stride
  const float* base  = feats + (size_t)b * kC * plane;  // batch base
  const int    off0  = y0 * kW + xt;
  const int    off1  = y1 * kW + xt;

  float* outp = out + (size_t)p * kC;

#pragma unroll
  for (int it = 0; it < kC / 32; ++it) {
    const int c0 = it * 32 + cl;        // channels for WMMA #0
    const int c1 = c0 + 16;             // channels for WMMA #1
    const float* pc0 = base + (size_t)c0 * plane;
    const float* pc1 = base + (size_t)c1 * plane;

    v2f B0, B1;                          // 4 independent gathers (latency overlap)
    B0.x = pc0[off0];  B0.y = pc0[off1];
    B1.x = pc1[off0];  B1.y = pc1[off1];

    float v;
#if HAS_WMMA_F32_4
    v8f acc0 = {};
    v8f acc1 = {};
    // D = A x B + 0. All rows of A identical => every row of D is the answer.
    // D v0: lanes 0-15 = row0 cols 0-15, lanes 16-31 = row8 cols 0-15.
    acc0 = __builtin_amdgcn_wmma_f32_16x16x4_f32(
        /*neg_a=*/false, A, /*neg_b=*/false, B0,
        /*c_mod=*/(short)0, acc0, /*reuse_a=*/false, /*reuse_b=*/false);
    acc1 = __builtin_amdgcn_wmma_f32_16x16x4_f32(
        false, A, false, B1, (short)0, acc1, false, false);
    // lanes 0-15 take channel c0 (from acc0), lanes 16-31 take c0+16 (from acc1):
    // both equal it*32 + lane  =>  fully coalesced 128B store from all 32 lanes.
    v = half ? acc1[0] : acc0[0];
#else
    // Fallback: each half holds 2 of the 4 taps; exchange partial sums.
    float s0 = A.x * B0.x + A.y * B0.y;
    float s1 = A.x * B1.x + A.y * B1.y;
    s0 += __shfl_xor(s0, 16, 32);
    s1 += __shfl_xor(s1, 16, 32);
    v = half ? s1 : s0;
#endif
    outp[it * 32 + lane] = v;
  }
}

extern "C" void kernel_launch(void* const* d_in, const int* in_sizes, int n_in,
                              void* d_out, int out_size, void* d_ws, size_t ws_size,
                              hipStream_t stream) {
  const float* feats   = (const float*)d_in[0];  // (4,256,188,188) fp32
  const float* centers = (const float*)d_in[1];  // (4,500,3) fp32
  float*       out     = (float*)d_out;          // (4,500,256) fp32

  // One wave per point, 8 waves (256 threads) per block: 2000/8 = 250 blocks.
  const int nblocks = (kB * kN) / 8;
  bev_bilinear_wmma<<<nblocks, 256, 0, stream>>>(feats, centers, out);
}